// GATv2_40510131535943
// MI455X (gfx1250) — compile-verified
//
#include <hip/hip_runtime.h>

#define N_NODES   50000
#define F_DIM     64
#define H_HEADS   2
#define HF        128          // H*F
#define L_LAYERS  2
#define NEG_SLOPE 0.2f

typedef __attribute__((ext_vector_type(2))) float v2f;
typedef __attribute__((ext_vector_type(8))) float v8f;

__device__ __forceinline__ float lrelu(float x) {
    return x > 0.0f ? x : NEG_SLOPE * x;
}
// order-preserving float<->uint flip so unsigned atomicMax == float max
__device__ __forceinline__ unsigned fflip(float f) {
    unsigned u = __float_as_uint(f);
    return (u & 0x80000000u) ? ~u : (u | 0x80000000u);
}
__device__ __forceinline__ float funflip(unsigned v) {
    return __uint_as_float((v & 0x80000000u) ? (v ^ 0x80000000u) : ~v);
}

// ---------------------------------------------------------------------------
// Fused projection GEMM: fs = h@Wsrc+bsrc, fd = h@Wdst+bdst  (WMMA f32 16x16x4)
// One wave per 16x16 output tile. grid=(3125,2), block=256 (8 waves).
// Col tiles 0..7 -> fs (Wsrc), 8..15 -> fd (Wdst).
// ---------------------------------------------------------------------------
__global__ __launch_bounds__(256)
void gat_proj_wmma(const float* __restrict__ h,
                   const float* __restrict__ Wsrc, const float* __restrict__ bsrc,
                   const float* __restrict__ Wdst, const float* __restrict__ bdst,
                   float* __restrict__ fs, float* __restrict__ fd) {
    const int wave = threadIdx.x >> 5;
    const int lane = threadIdx.x & 31;
    const int half = lane >> 4;        // 0: lanes 0-15, 1: lanes 16-31
    const int lrow = lane & 15;
    const int t = blockIdx.y * 8 + wave;          // column tile 0..15
    const int nodeBase = blockIdx.x * 16;         // 3125*16 == 50000 exactly

    const bool  isSrc = (t < 8);
    const float* __restrict__ W    = isSrc ? Wsrc : Wdst;
    const float* __restrict__ bias = isSrc ? bsrc : bdst;
    float* __restrict__ out        = isSrc ? fs : fd;
    const int col = (isSrc ? t : t - 8) * 16 + lrow;   // 0..127 within matrix

    // A-operand row for this lane (ISA 16x4 f32 layout)
    const float* __restrict__ arow = h + (size_t)(nodeBase + lrow) * F_DIM;

    v8f acc = {};
#pragma unroll
    for (int k0 = 0; k0 < F_DIM; k0 += 4) {
        const int ka = k0 + half * 2;            // K index pair for this lane
        v2f a, b;
        a.x = arow[ka];
        a.y = arow[ka + 1];
        b.x = W[(size_t)ka * HF + col];          // B[K][N], K striped like A
        b.y = W[(size_t)(ka + 1) * HF + col];
        // 8 args: (neg_a, A, neg_b, B, c_mod, C, reuse_a, reuse_b)
        acc = __builtin_amdgcn_wmma_f32_16x16x4_f32(
                false, a, false, b, (short)0, acc, false, false);
    }

    const float bv = bias[col];
#pragma unroll
    for (int j = 0; j < 8; ++j) {               // D layout: VGPR j = row j + half*8
        const int row = nodeBase + half * 8 + j;
        out[(size_t)row * HF + col] = acc[j] + bv;
    }
}

// ---------------------------------------------------------------------------
// Per-edge GATv2 logits + running segment-max. One wave per edge.
// lanes 0-15 : head 0 (feat 0..63), lanes 16-31 : head 1 (feat 64..127).
// ---------------------------------------------------------------------------
__global__ __launch_bounds__(256)
void gat_edge_logits(const float* __restrict__ fs, const float* __restrict__ fd,
                     const float* __restrict__ attn,
                     const int* __restrict__ src, const int* __restrict__ dst,
                     float* __restrict__ logits, unsigned* __restrict__ mflip,
                     int E) {
    const int e = (int)(((size_t)blockIdx.x * blockDim.x + threadIdx.x) >> 5);
    if (e >= E) return;                          // wave-uniform exit
    const int lane = threadIdx.x & 31;
    const int s = src[e];
    const int d = dst[e];

    const float4 a = *(const float4*)(fs + (size_t)s * HF + lane * 4);
    const float4 b = *(const float4*)(fd + (size_t)d * HF + lane * 4);
    const float4 w = *(const float4*)(attn + lane * 4);

    float sum = w.x * lrelu(a.x + b.x) + w.y * lrelu(a.y + b.y)
              + w.z * lrelu(a.z + b.z) + w.w * lrelu(a.w + b.w);
#pragma unroll
    for (int off = 1; off < 16; off <<= 1)       // reduce within each 16-lane head
        sum += __shfl_xor(sum, off, 32);

    if ((lane & 15) == 0) {
        const int hh = lane >> 4;
        logits[(size_t)e * 2 + hh] = sum;
        atomicMax(mflip + (size_t)d * 2 + hh, fflip(sum));
    }
}

// ---------------------------------------------------------------------------
// ex = exp(logit - max[dst]); denom[dst] += ex. One thread per (edge, head).
// ---------------------------------------------------------------------------
__global__ __launch_bounds__(256)
void gat_edge_exp(const int* __restrict__ dst, float* __restrict__ logits,
                  const unsigned* __restrict__ mflip, float* __restrict__ denom,
                  int E2) {
    const int idx = blockIdx.x * blockDim.x + threadIdx.x;
    if (idx >= E2) return;
    const int e = idx >> 1, hh = idx & 1;
    const int d = dst[e];
    const float m  = funflip(mflip[(size_t)d * 2 + hh]);
    const float ex = expf(logits[idx] - m);
    logits[idx] = ex;
    atomicAdd(denom + (size_t)d * 2 + hh, ex);
}

// ---------------------------------------------------------------------------
// agg[dst] += (ex/denom[dst]) * fs[src]. One wave per edge, float4 per lane.
// ---------------------------------------------------------------------------
__global__ __launch_bounds__(256)
void gat_edge_aggregate(const float* __restrict__ fs,
                        const int* __restrict__ src, const int* __restrict__ dst,
                        const float* __restrict__ logits,
                        const float* __restrict__ denom,
                        float* __restrict__ agg, int E) {
    const int e = (int)(((size_t)blockIdx.x * blockDim.x + threadIdx.x) >> 5);
    if (e >= E) return;
    const int lane = threadIdx.x & 31;
    const int hh = lane >> 4;
    const int s = src[e];
    const int d = dst[e];
    const float alpha = logits[(size_t)e * 2 + hh] / denom[(size_t)d * 2 + hh];

    const float4 v = *(const float4*)(fs + (size_t)s * HF + lane * 4);
    float* o = agg + (size_t)d * HF + lane * 4;
    atomicAdd(o + 0, alpha * v.x);
    atomicAdd(o + 1, alpha * v.y);
    atomicAdd(o + 2, alpha * v.z);
    atomicAdd(o + 3, alpha * v.w);
}

// ---------------------------------------------------------------------------
// Head mean: hout[n][f] = 0.5*(agg[n][0][f] + agg[n][1][f])
// ---------------------------------------------------------------------------
__global__ __launch_bounds__(256)
void gat_head_mean(const float* __restrict__ agg, float* __restrict__ hout, int n) {
    const int idx = blockIdx.x * blockDim.x + threadIdx.x;
    if (idx >= n) return;
    const int node = idx >> 6, f = idx & 63;
    hout[idx] = 0.5f * (agg[(size_t)node * HF + f] + agg[(size_t)node * HF + 64 + f]);
}

// ---------------------------------------------------------------------------
extern "C" void kernel_launch(void* const* d_in, const int* in_sizes, int n_in,
                              void* d_out, int out_size, void* d_ws, size_t ws_size,
                              hipStream_t stream) {
    (void)n_in; (void)out_size; (void)ws_size;
    const float* h0   = (const float*)d_in[0];
    const float* Wsrc = (const float*)d_in[1];
    const float* bsrc = (const float*)d_in[2];
    const float* Wdst = (const float*)d_in[3];
    const float* bdst = (const float*)d_in[4];
    const float* attn = (const float*)d_in[5];
    const int*   src  = (const int*)d_in[6];
    const int*   dst  = (const int*)d_in[7];
    const int    E    = in_sizes[6];

    // workspace carve-up (floats)
    float* ws      = (float*)d_ws;
    float* fs      = ws;  ws += (size_t)N_NODES * HF;
    float* fd      = ws;  ws += (size_t)N_NODES * HF;
    float* agg     = ws;  ws += (size_t)N_NODES * HF;
    float* hbuf    = ws;  ws += (size_t)N_NODES * F_DIM;
    float* denom   = ws;  ws += (size_t)N_NODES * 2;
    unsigned* mflp = (unsigned*)ws; ws += (size_t)N_NODES * 2;
    float* logits  = ws;  // E*2 floats

    const int gridE  = (E + 7) / 8;                 // 8 waves (edges) per block
    const int gridE2 = (E * 2 + 255) / 256;
    const int gridN  = (N_NODES * F_DIM + 255) / 256;

    const float* hcur = h0;
    for (int l = 0; l < L_LAYERS; ++l) {
        gat_proj_wmma<<<dim3(N_NODES / 16, 2), 256, 0, stream>>>(
            hcur,
            Wsrc + (size_t)l * F_DIM * HF, bsrc + (size_t)l * HF,
            Wdst + (size_t)l * F_DIM * HF, bdst + (size_t)l * HF,
            fs, fd);

        hipMemsetAsync(mflp,  0, (size_t)N_NODES * 2 * sizeof(unsigned), stream);
        hipMemsetAsync(denom, 0, (size_t)N_NODES * 2 * sizeof(float), stream);
        hipMemsetAsync(agg,   0, (size_t)N_NODES * HF * sizeof(float), stream);

        gat_edge_logits<<<gridE, 256, 0, stream>>>(
            fs, fd, attn + (size_t)l * HF, src, dst, logits, mflp, E);
        gat_edge_exp<<<gridE2, 256, 0, stream>>>(dst, logits, mflp, denom, E * 2);
        gat_edge_aggregate<<<gridE, 256, 0, stream>>>(
            fs, src, dst, logits, denom, agg, E);

        float* hn = (l == L_LAYERS - 1) ? (float*)d_out : hbuf;
        gat_head_mean<<<gridN, 256, 0, stream>>>(agg, hn, N_NODES * F_DIM);
        hcur = hbuf;
    }
}